// DeformableAlignment_66271345377652
// MI455X (gfx1250) — compile-verified
//
#include <hip/hip_runtime.h>

typedef __attribute__((ext_vector_type(2))) float v2f;
typedef __attribute__((ext_vector_type(8))) float v8f;

#define B_    2
#define H_    128
#define W_    128
#define HW_   (H_*W_)
#define C_    64
#define OFFC_ 432
#define MRM_  10.0f

__device__ __forceinline__ float leaky01(float v) { return v >= 0.f ? v : 0.1f * v; }

// LDS byte offset of a pointer into a __shared__ array (ptrtoint of AS(3) pointer).
__device__ __forceinline__ unsigned ldsOff(void* p) {
    return (unsigned)(unsigned long long)(__attribute__((address_space(3))) char*)p;
}

// Async global->LDS copy, 4 bytes per lane (GLOBAL_LOAD_ASYNC_TO_LDS_B32, ASYNCcnt).
__device__ __forceinline__ void asyncCopyB32(unsigned lds_byte_off, const float* gsrc) {
    asm volatile("global_load_async_to_lds_b32 %0, %1, off"
                 :: "v"(lds_byte_off), "v"(gsrc) : "memory");
}

__device__ __forceinline__ void waitAsync0() {
    asm volatile("s_wait_asynccnt 0x0" ::: "memory");
}

// Implicit-GEMM 3x3 conv, pad=1, stride=1.
// Block: 128 threads (4 wave32). Tile: 16 out-channels (blockIdx.y) x 64 pixels (blockIdx.x).
// K order per 16-channel chunk: k' = kk*16 + ci (tap-major) so WMMA fragment addresses are
// affine with compile-time immediates -> ds_load_2addr, no div/mod in the hot loop.
template<int CIN, int COUT, bool LEAKY>
__global__ __launch_bounds__(128) void conv3x3_wmma(const float* __restrict__ in,
                                                    const float* __restrict__ w,
                                                    const float* __restrict__ bias,
                                                    float* __restrict__ out)
{
    __shared__ float InLds[16][3][68];   // 16 cin x 3 rows x (64+2 halo) cols (padded)
    __shared__ float WLds[16][144];      // 16 outch x (9 taps * 16 cin), k' = kk*16+ci

    const int tid  = threadIdx.x;
    const int lane = tid & 31;
    const int wid  = tid >> 5;
    const int b    = blockIdx.z;
    const int mo   = blockIdx.y * 16;
    const int pb   = blockIdx.x;
    const int y    = pb >> 1;            // H=128, two 64-pixel segments per row
    const int x0   = (pb & 1) * 64;

    const int mrow  = lane & 15;
    const int khalf = (lane >> 4) * 2;   // lanes 0-15 -> K {0,1}; 16-31 -> K {2,3}
    const int p     = wid * 16 + (lane & 15);   // pixel within the 64-pixel block

    v8f acc = {};

    for (int c0 = 0; c0 < CIN; c0 += 16) {
        // --- async global->LDS fill of input slab (zero-stored halo) ---
        for (int idx = tid; idx < 16 * 3 * 66; idx += 128) {
            int ck  = idx / 198;
            int rem = idx - ck * 198;
            int r   = rem / 66;
            int col = rem - r * 66;
            int yy  = y - 1 + r;
            int xx  = x0 - 1 + col;
            if (yy >= 0 && yy < H_ && xx >= 0 && xx < W_) {
                asyncCopyB32(ldsOff(&InLds[ck][r][col]),
                             &in[(((size_t)b * CIN + c0 + ck) * H_ + yy) * W_ + xx]);
            } else {
                InLds[ck][r][col] = 0.f;
            }
        }
        // --- async global->LDS fill of weight chunk, permuted to k' = kk*16+ci ---
        for (int idx = tid; idx < 16 * 144; idx += 128) {
            int m  = idx / 144;
            int kp = idx - m * 144;
            int kk = kp >> 4;
            int ci = kp & 15;
            asyncCopyB32(ldsOff(&WLds[m][kp]),
                         &w[(((size_t)(mo + m)) * CIN + (c0 + ci)) * 9 + kk]);
        }
        waitAsync0();
        __syncthreads();

        // --- 36 WMMAs, fully unrolled: all LDS addresses are base + immediates ---
        #pragma unroll
        for (int kk = 0; kk < 9; ++kk) {
            const int ky = kk / 3;
            const int kx = kk - ky * 3;
            #pragma unroll
            for (int c4 = 0; c4 < 16; c4 += 4) {
                const int kA = kk * 16 + c4 + khalf;
                v2f a, bb;
                a.x  = WLds[mrow][kA];
                a.y  = WLds[mrow][kA + 1];
                bb.x = InLds[c4 + khalf][ky][p + kx];
                bb.y = InLds[c4 + khalf + 1][ky][p + kx];
                acc = __builtin_amdgcn_wmma_f32_16x16x4_f32(false, a, false, bb,
                                                            (short)0, acc, false, false);
            }
        }
        __syncthreads();
    }

    // D layout: VGPR r -> M=r (lanes 0-15) / M=8+r (lanes 16-31); N = lane&15
    const int mloc = (lane >> 4) * 8;
    #pragma unroll
    for (int r = 0; r < 8; ++r) {
        int m   = mo + mloc + r;
        float v = acc[r] + bias[m];
        if (LEAKY) v = leaky01(v);
        out[(((size_t)b * COUT + m) * H_ + y) * W_ + x0 + p] = v;
    }
}

// Deformable bilinear gather + final GEMM (K=576, M=64, N=16 pixels per block).
// Block: 128 threads (4 waves); each wave owns a 16-outch M tile.
__global__ __launch_bounds__(128) void deform_wmma(const float* __restrict__ x,
                                                   const float* __restrict__ flow,
                                                   const float* __restrict__ o,
                                                   const float* __restrict__ weight,
                                                   float* __restrict__ out)
{
    __shared__ float valLds[576][16];    // K x pixels, 36 KB

    const int tid  = threadIdx.x;
    const int lane = tid & 31;
    const int wid  = tid >> 5;

    const int q0    = blockIdx.x * 16;
    const int b     = q0 / HW_;
    const int rem   = q0 - b * HW_;
    const int y     = rem / W_;
    const int xbase = rem - y * W_;      // 16-aligned, same row for all 16 pixels

    const size_t ob   = (size_t)b * OFFC_ * HW_;
    const size_t fb   = (size_t)b * 2 * HW_;
    const size_t spix = (size_t)y * W_;

    // Phase 1: offsets (tanh), mask (sigmoid), masked bilinear sampling into LDS.
    for (int item = tid; item < 16 * 16 * 9; item += 128) {
        int j  = item & 15;
        int gk = item >> 4;
        int g  = gk / 9;
        int kk = gk - g * 9;
        int xx = xbase + j;
        size_t pix = spix + xx;
        int chy = g * 18 + kk * 2;
        float offy = MRM_ * tanhf(o[ob + (size_t)chy * HW_ + pix])       + flow[fb + HW_ + pix];
        float offx = MRM_ * tanhf(o[ob + (size_t)(chy + 1) * HW_ + pix]) + flow[fb + pix];
        float mval = o[ob + (size_t)(288 + g * 9 + kk) * HW_ + pix];
        float mask = 1.f / (1.f + __expf(-mval));
        float py = (float)y  + (float)(kk / 3 - 1) + offy;
        float px = (float)xx + (float)(kk % 3 - 1) + offx;
        float y0f = floorf(py), x0f = floorf(px);
        float wy = py - y0f, wx = px - x0f;
        int y0 = (int)y0f, x0i = (int)x0f;
        int y1 = y0 + 1, x1 = x0i + 1;
        float w00 = (1.f - wy) * (1.f - wx), w01 = (1.f - wy) * wx;
        float w10 = wy * (1.f - wx),         w11 = wy * wx;
        bool vy0 = (y0 >= 0) && (y0 < H_);
        bool vy1 = (y1 >= 0) && (y1 < H_);
        bool vx0 = (x0i >= 0) && (x0i < W_);
        bool vx1 = (x1 >= 0) && (x1 < W_);
        #pragma unroll
        for (int c = 0; c < 4; ++c) {
            int cin = g * 4 + c;
            const float* xb = x + ((size_t)b * C_ + cin) * HW_;
            float v = 0.f;
            if (vy0 && vx0) v += w00 * xb[y0 * W_ + x0i];
            if (vy0 && vx1) v += w01 * xb[y0 * W_ + x1];
            if (vy1 && vx0) v += w10 * xb[y1 * W_ + x0i];
            if (vy1 && vx1) v += w11 * xb[y1 * W_ + x1];
            valLds[(g * 4 + c) * 9 + kk][j] = v * mask;
        }
    }
    __syncthreads();

    // Phase 2: 144 f32 WMMAs; weight (64,64,3,3) is already [O][576] in K order.
    const int mo    = wid * 16;
    const int mrow  = mo + (lane & 15);
    const int khalf = (lane >> 4) * 2;
    const float* wrow = weight + (size_t)mrow * 576;
    __builtin_prefetch(wrow, 0, 0);      // global_prefetch_b8: warm the weight row
    v8f acc = {};
    #pragma unroll 4
    for (int k0 = 0; k0 < 576; k0 += 4) {
        v2f a, bb;
        a.x  = wrow[k0 + khalf];
        a.y  = wrow[k0 + khalf + 1];
        bb.x = valLds[k0 + khalf][lane & 15];
        bb.y = valLds[k0 + khalf + 1][lane & 15];
        acc = __builtin_amdgcn_wmma_f32_16x16x4_f32(false, a, false, bb,
                                                    (short)0, acc, false, false);
    }

    const int mloc = (lane >> 4) * 8;
    #pragma unroll
    for (int r = 0; r < 8; ++r) {
        int m = mo + mloc + r;
        out[(((size_t)b * C_ + m) * H_ + y) * W_ + xbase + (lane & 15)] = acc[r];
    }
}

extern "C" void kernel_launch(void* const* d_in, const int* in_sizes, int n_in,
                              void* d_out, int out_size, void* d_ws, size_t ws_size,
                              hipStream_t stream) {
    const float* x     = (const float*)d_in[0];
    const float* extra = (const float*)d_in[1];
    const float* flow  = (const float*)d_in[2];
    const float* w1    = (const float*)d_in[3];
    const float* b1    = (const float*)d_in[4];
    const float* w2    = (const float*)d_in[5];
    const float* b2    = (const float*)d_in[6];
    const float* w3    = (const float*)d_in[7];
    const float* b3    = (const float*)d_in[8];
    const float* wgt   = (const float*)d_in[9];
    float* out = (float*)d_out;

    // Workspace layout (floats): h1 (8MB), h2 (8MB), offsets o (56.6MB)
    float* h1 = (float*)d_ws;
    float* h2 = h1 + (size_t)B_ * C_ * HW_;
    float* oo = h2 + (size_t)B_ * C_ * HW_;

    dim3 blk(128, 1, 1);
    conv3x3_wmma<128,  64, true ><<<dim3(256,  4, B_), blk, 0, stream>>>(extra, w1, b1, h1);
    conv3x3_wmma< 64,  64, true ><<<dim3(256,  4, B_), blk, 0, stream>>>(h1,    w2, b2, h2);
    conv3x3_wmma< 64, 432, false><<<dim3(256, 27, B_), blk, 0, stream>>>(h2,    w3, b3, oo);
    deform_wmma<<<dim3((B_ * HW_) / 16, 1, 1), blk, 0, stream>>>(x, flow, oo, wgt, out);
}